// YOLOloss_56203942035516
// MI455X (gfx1250) — compile-verified
//
#include <hip/hip_runtime.h>

typedef __attribute__((ext_vector_type(2))) float v2f;
typedef __attribute__((ext_vector_type(8))) float v8f;

#define NUM_B 32
#define NUM_M 20
#define NB3 256
#define NB4 64
#define NB5 16
// d_ws float layout:
// [0,336)   conf noobj partials (256 p3 + 64 p4 + 16 p5)
// [336,432) coord partials (96 obj blocks)
// [432,528) conf adjustment partials (96)
// [528,624) cls partials (96)

__device__ const float c_anch[3][3][2] = {
    {{10.f, 13.f}, {16.f, 30.f}, {33.f, 23.f}},
    {{30.f, 61.f}, {62.f, 45.f}, {59.f, 119.f}},
    {{116.f, 90.f}, {156.f, 198.f}, {373.f, 326.f}}};

__device__ __forceinline__ float sigmoidf_(float z) {
  return 1.0f / (1.0f + __expf(-z));
}

// min(softplus(z), 100) == clamped -log(1 - sigmoid(z)); sp(-z) gives -log(sigmoid(z))
__device__ __forceinline__ float sp_clamp(float z) {
  float sp = fmaxf(z, 0.0f) + log1pf(__expf(-fabsf(z)));
  return fminf(sp, 100.0f);
}

// Sum (a0 + a1) over all 32 lanes using the matrix unit:
// A(16x4) holds the 64 partials, B = ones -> D rows are 4-element sums;
// 8 adds + one xor-16 shuffle completes the reduction. All lanes get the total.
__device__ __forceinline__ float wave_reduce32(float a0, float a1) {
  v2f A; A.x = a0; A.y = a1;
  v2f Bv; Bv.x = 1.0f; Bv.y = 1.0f;
  v8f C = {0.f, 0.f, 0.f, 0.f, 0.f, 0.f, 0.f, 0.f};
  v8f D = __builtin_amdgcn_wmma_f32_16x16x4_f32(false, A, false, Bv,
                                                (short)0, C, false, false);
  float s = D[0] + D[1] + D[2] + D[3] + D[4] + D[5] + D[6] + D[7];
  s += __shfl_xor(s, 16, 32);
  return s;
}

template <int G>
__device__ __forceinline__ float conf_noobj(const float* __restrict__ pred, int e) {
  constexpr int GG = G * G;
  int p = e / GG;          // p = b*3 + a
  int cell = e - p * GG;
  int b = p / 3;
  int a = p - b * 3;
  int off = (b * 255 + a * 85 + 4) * GG + cell;  // conf channel plane
  return 0.5f * sp_clamp(pred[off]);
}

template <int G, int NB>
__global__ __launch_bounds__(256) void conf_kernel(const float* __restrict__ pred,
                                                   float* __restrict__ partials) {
  constexpr int GG = G * G;
  constexpr int N = NUM_B * 3 * GG;
  const int T = NB * 256;
  int t = blockIdx.x * 256 + threadIdx.x;
  float a0 = 0.f, a1 = 0.f;  // two accumulators feed the WMMA A operand
  for (int e = t; e < N; e += 2 * T) {
    a0 += conf_noobj<G>(pred, e);
    int e2 = e + T;
    if (e2 < N) a1 += conf_noobj<G>(pred, e2);
  }
  float s = wave_reduce32(a0, a1);
  __shared__ float sw[8];
  int lane = threadIdx.x & 31, wid = threadIdx.x >> 5;
  if (lane == 0) sw[wid] = s;
  __syncthreads();
  if (threadIdx.x == 0) {
    float tot = 0.f;
#pragma unroll
    for (int i = 0; i < 8; ++i) tot += sw[i];
    partials[blockIdx.x] = tot;
  }
}

__global__ __launch_bounds__(32) void obj_kernel(
    const float* __restrict__ p3, const float* __restrict__ p4,
    const float* __restrict__ p5, const float* __restrict__ boxes,
    const int* __restrict__ labels, float* __restrict__ coordP,
    float* __restrict__ confP, float* __restrict__ clsP) {
  int blk = blockIdx.x;  // 0..95
  int s = blk / NUM_B;
  int b = blk - s * NUM_B;
  const float* pred;
  int G;
  float stride;
  if (s == 0)      { pred = p3; G = 52; stride = 8.f; }
  else if (s == 1) { pred = p4; G = 26; stride = 16.f; }
  else             { pred = p5; G = 13; stride = 32.f; }
  int GG = G * G;

  __shared__ int sKey[NUM_M];
  __shared__ int sLab[NUM_M];
  __shared__ float sT[4][NUM_M];

  int lane = threadIdx.x;
  if (lane < NUM_M) {
    const float* bx = boxes + (b * NUM_M + lane) * 4;
    float gx = bx[0] * 416.f, gy = bx[1] * 416.f;
    float gw = bx[2] * 416.f, gh = bx[3] * 416.f;
    float inv = 1.0f / stride;
    float fx = gx * inv, fy = gy * inv;
    int gi = (int)floorf(fx), gj = (int)floorf(fy);
    float bestIou = -1.f;
    int best = 0;
    float bw = 1.f, bh = 1.f;
#pragma unroll
    for (int a = 0; a < 3; ++a) {
      float aw = c_anch[s][a][0] * stride, ah = c_anch[s][a][1] * stride;
      float inter = fminf(gw, aw) * fminf(gh, ah);
      float uni = gw * gh + aw * ah - inter;
      float iou = inter / uni;
      if (iou > bestIou) { bestIou = iou; best = a; bw = aw; bh = ah; }
    }
    bool valid = (gi >= 0) && (gi < G) && (gj >= 0) && (gj < G);
    sKey[lane] = valid ? ((best << 16) | (gj << 8) | gi) : -(lane + 1);
    sT[0][lane] = fx - (float)gi;
    sT[1][lane] = fy - (float)gj;
    sT[2][lane] = logf(gw / bw + 1e-16f);
    sT[3][lane] = logf(gh / bh + 1e-16f);
    sLab[lane] = labels[b * NUM_M + lane];
  }
  __syncthreads();

  float coordA = 0.f, confA = 0.f, clsA = 0.f;
  for (int m = 0; m < NUM_M; ++m) {        // wave-uniform control flow
    int key = sKey[m];
    if (key < 0) continue;                 // dropped (out-of-grid) box
    bool owner = true;                     // scatter-set: last writer wins
    for (int mm = m + 1; mm < NUM_M; ++mm)
      if (sKey[mm] == key) { owner = false; break; }
    if (!owner) continue;
    unsigned mk0 = 0, mk1 = 0, mk2 = 0;    // union of labels hitting this cell
    for (int mm = 0; mm < NUM_M; ++mm) {
      if (sKey[mm] == key) {
        int L = sLab[mm];
        if (L < 32) mk0 |= 1u << L;
        else if (L < 64) mk1 |= 1u << (L - 32);
        else mk2 |= 1u << (L - 64);
      }
    }
    int best = key >> 16, gj = (key >> 8) & 255, gi = key & 255;
    int base = (b * 255 + best * 85) * GG + gj * G + gi;
#pragma unroll
    for (int k = 0; k < 3; ++k) {          // 32 lanes cover 85 channels
      int c = lane + 32 * k;
      if (c < 85) {
        float z = pred[base + c * GG];
        if (c == 0)      { float d = sigmoidf_(z) - sT[0][m]; coordA += d * d; }
        else if (c == 1) { float d = sigmoidf_(z) - sT[1][m]; coordA += d * d; }
        else if (c == 2) { float d = z - sT[2][m]; coordA += d * d; }
        else if (c == 3) { float d = z - sT[3][m]; coordA += d * d; }
        else if (c == 4) {
          // obj conf term, minus the noobj term the big pass counted here
          confA += sp_clamp(-z) - 0.5f * sp_clamp(z);
        } else {
          int L = c - 5;
          unsigned t = (L < 32) ? (mk0 >> L)
                     : (L < 64) ? (mk1 >> (L - 32))
                                : (mk2 >> (L - 64));
          clsA += (t & 1u) ? sp_clamp(-z) : sp_clamp(z);
        }
      }
    }
  }
  float cS = wave_reduce32(coordA, 0.f);
  float fS = wave_reduce32(confA, 0.f);
  float lS = wave_reduce32(clsA, 0.f);
  if (lane == 0) { coordP[blk] = cS; confP[blk] = fS; clsP[blk] = lS; }
}

__global__ __launch_bounds__(32) void final_kernel(const float* __restrict__ ws,
                                                   float* __restrict__ out) {
  int lane = threadIdx.x;
  float confA = 0.f, coordA = 0.f, clsA = 0.f;
  for (int i = lane; i < 336; i += 32) confA += ws[i];
  for (int i = lane; i < 96; i += 32) {
    coordA += ws[336 + i];
    confA  += ws[432 + i];
    clsA   += ws[528 + i];
  }
  float conf  = wave_reduce32(confA, 0.f) * (1.0f / NUM_B);
  float coord = 5.0f * wave_reduce32(coordA, 0.f) * (1.0f / NUM_B);
  float cls   = wave_reduce32(clsA, 0.f) * (1.0f / NUM_B);
  if (lane == 0) {
    out[0] = coord + conf + cls;
    out[1] = coord;
    out[2] = conf;
    out[3] = cls;
  }
}

extern "C" void kernel_launch(void* const* d_in, const int* in_sizes, int n_in,
                              void* d_out, int out_size, void* d_ws,
                              size_t ws_size, hipStream_t stream) {
  const float* p3 = (const float*)d_in[0];
  const float* p4 = (const float*)d_in[1];
  const float* p5 = (const float*)d_in[2];
  const float* boxes = (const float*)d_in[3];
  const int* labels = (const int*)d_in[4];
  float* ws = (float*)d_ws;
  float* out = (float*)d_out;

  conf_kernel<52, NB3><<<NB3, 256, 0, stream>>>(p3, ws + 0);
  conf_kernel<26, NB4><<<NB4, 256, 0, stream>>>(p4, ws + NB3);
  conf_kernel<13, NB5><<<NB5, 256, 0, stream>>>(p5, ws + NB3 + NB4);
  obj_kernel<<<96, 32, 0, stream>>>(p3, p4, p5, boxes, labels,
                                    ws + 336, ws + 432, ws + 528);
  final_kernel<<<1, 32, 0, stream>>>(ws, out);
}